// SpatialAttention_1228360647387
// MI455X (gfx1250) — compile-verified
//
#include <hip/hip_runtime.h>
#include <math.h>

typedef float v2f __attribute__((ext_vector_type(2)));
typedef float v8f __attribute__((ext_vector_type(8)));

#define NN 64
#define CC 128
#define TT 64
#define VV 50
#define SEGN 16
#define BB 256                   // 4*N segment batches
#define CV (CC * VV)             // 6400
#define TOT (NN * CC * TT * VV)  // 26214400

// -------- workspace layout (in floats) --------
#define O_SUMT 0                       // [BB*16]  sum over (c,v) per (b,s)
#define O_MAXT (O_SUMT + BB * SEGN)    // [BB*16]  max over (c,v)
#define O_SUMC (O_MAXT + BB * SEGN)    // [BB*16*VV] sum over c
#define O_MAXC (O_SUMC + BB * SEGN * VV)
#define O_MINC (O_MAXC + BB * SEGN * VV)
#define O_OUTT (O_MINC + BB * SEGN * VV)  // [BB*16] temporal gate
#define O_G    (O_OUTT + BB * SEGN)       // [BB*VV] spatial gate
#define O_AMAT (O_G + BB * VV)            // [3*VV*VV]

// =====================================================================
// Kernel 1: pooling pass — single read of x.
// grid = BB*16 blocks, 256 threads. Per (b,s): total sum/max over (c,v),
// plus per-v sum/max/min over c (needed for signed-gate spatial pooling).
// =====================================================================
__global__ void __launch_bounds__(256) pool_kernel(
    const float* __restrict__ x, float* __restrict__ ws) {
  const int bs = blockIdx.x;        // 0..4095
  const int b = bs >> 4, s = bs & 15;
  const int g = b >> 6, n = b & 63;
  const int t = g * SEGN + s;
  const float* base = x + ((size_t)n * CC * TT + t) * VV;  // c-stride = TT*VV

  const int tid = threadIdx.x;
  const int v = tid & 63, cg = tid >> 6;  // 4 c-groups of 32 channels

  __shared__ float sS[4 * 64], sMx[4 * 64], sMn[4 * 64];

  float acc = 0.f, mx = -__builtin_inff(), mn = __builtin_inff();
  if (v < VV) {
    const float* p = base + (size_t)(cg * 32) * (TT * VV) + v;
    #pragma unroll 4
    for (int i = 0; i < 32; ++i) {
      float val = *p;
      p += TT * VV;
      acc += val;
      mx = fmaxf(mx, val);
      mn = fminf(mn, val);
    }
  }
  sS[cg * 64 + v] = acc;
  sMx[cg * 64 + v] = mx;
  sMn[cg * 64 + v] = mn;
  __syncthreads();

  if (tid < VV) {
    float a0 = sS[tid] + sS[64 + tid] + sS[128 + tid] + sS[192 + tid];
    float m0 = fmaxf(fmaxf(sMx[tid], sMx[64 + tid]),
                     fmaxf(sMx[128 + tid], sMx[192 + tid]));
    float n0 = fminf(fminf(sMn[tid], sMn[64 + tid]),
                     fminf(sMn[128 + tid], sMn[192 + tid]));
    const size_t o = (size_t)bs * VV + tid;
    ws[O_SUMC + o] = a0;
    ws[O_MAXC + o] = m0;
    ws[O_MINC + o] = n0;
    sS[tid] = a0;
    sMx[tid] = m0;
  }
  __syncthreads();
  if (tid == 0) {
    float a = 0.f, m = -__builtin_inff();
    for (int i = 0; i < VV; ++i) {
      a += sS[i];
      m = fmaxf(m, sMx[i]);
    }
    ws[O_SUMT + bs] = a;
    ws[O_MAXT + bs] = m;
  }
}

// =====================================================================
// Kernel 2: temporal MLP via f32 WMMA. fcT is (256x16)@(16x16) twice, for
// avg and max pooled inputs: exactly chains of V_WMMA_F32_16X16X4_F32.
// A layout (ISA 7.12.2): lane = M, VGPR pair = K (lanes 16-31 hold K+2).
// D layout: VGPR r = row M=r(+8 for hi lanes), lane = N.
// =====================================================================
__device__ __forceinline__ v8f wmma_gemm16(const float* __restrict__ A,
                                           const float* __restrict__ W,  // (out,in) row-major; B[k][n]=W[n][k]
                                           v8f c) {
  const int lane = threadIdx.x & 31;
  const int m = lane & 15;
  const int hi2 = (lane >> 4) << 1;  // 0 or 2
  #pragma unroll
  for (int kb = 0; kb < 16; kb += 4) {
    v2f a, bm;
    a.x  = A[m * 16 + kb + hi2];
    a.y  = A[m * 16 + kb + hi2 + 1];
    bm.x = W[m * 16 + kb + hi2];      // n = m
    bm.y = W[m * 16 + kb + hi2 + 1];
    c = __builtin_amdgcn_wmma_f32_16x16x4_f32(false, a, false, bm,
                                              (short)0, c, false, false);
  }
  return c;
}

__global__ void __launch_bounds__(32) fcT_wmma_kernel(
    const float* __restrict__ ws_in,
    const float* __restrict__ WT1, const float* __restrict__ bT1,
    const float* __restrict__ WT2, const float* __restrict__ bT2,
    float* __restrict__ outT) {
  __shared__ float Za[256], Zm[256], H[256];
  const int lane = threadIdx.x;
  const int m0 = blockIdx.x * 16;  // 16 blocks cover 256 batch rows
  const float invCV = 1.0f / (float)CV;

  for (int i = lane; i < 256; i += 32) {
    const int row = i >> 4, k = i & 15;
    Za[i] = ws_in[O_SUMT + (m0 + row) * SEGN + k] * invCV;  // avg pool
    Zm[i] = ws_in[O_MAXT + (m0 + row) * SEGN + k];          // max pool
  }
  __syncthreads();

  const int m = lane & 15;
  const int hi = lane >> 4;
  v8f z;
  #pragma unroll
  for (int i = 0; i < 8; ++i) z[i] = 0.f;

  // ---- avg path: L1 -> relu+bias -> L2 ----
  v8f h = wmma_gemm16(Za, WT1, z);
  #pragma unroll
  for (int r = 0; r < 8; ++r)
    H[(r + 8 * hi) * 16 + m] = fmaxf(h[r] + bT1[m], 0.0f);
  __syncthreads();
  v8f oA = wmma_gemm16(H, WT2, z);
  __syncthreads();

  // ---- max path ----
  h = wmma_gemm16(Zm, WT1, z);
  #pragma unroll
  for (int r = 0; r < 8; ++r)
    H[(r + 8 * hi) * 16 + m] = fmaxf(h[r] + bT1[m], 0.0f);
  __syncthreads();
  v8f oM = wmma_gemm16(H, WT2, z);

  #pragma unroll
  for (int r = 0; r < 8; ++r) {
    const int row = m0 + r + 8 * hi;
    outT[row * SEGN + m] = 0.5f * (oA[r] + oM[r]) + bT2[m];
  }
}

// =====================================================================
// Kernel 3: a_mats. Since past_A==0, dif==A and fc_A is pointwise:
// amat = WA2 . relu((WA1[:,0]+WA1[:,1])*a + bA1) + bA2
// =====================================================================
__global__ void amat_kernel(const float* __restrict__ A,
                            const float* __restrict__ WA1,
                            const float* __restrict__ bA1,
                            const float* __restrict__ WA2,
                            const float* __restrict__ bA2,
                            float* __restrict__ amat) {
  const int idx = blockIdx.x * blockDim.x + threadIdx.x;
  if (idx >= 3 * VV * VV) return;
  const float a = A[idx];
  float out = bA2[0];
  #pragma unroll
  for (int j = 0; j < 9; ++j) {
    const float hh = fmaxf((WA1[j * 2] + WA1[j * 2 + 1]) * a + bA1[j], 0.0f);
    out += WA2[j] * hh;
  }
  amat[idx] = out;
}

// =====================================================================
// Kernel 4: per-b spatial attention + gate. Uses the per-(b,s,v) c-stats so
// no re-read of x is needed. max over (c,s) of outT[s]*x handled sign-aware.
// =====================================================================
__global__ void __launch_bounds__(64) spatial_kernel(
    const float* __restrict__ ws_in, const float* __restrict__ amat,
    const float* __restrict__ WS1, const float* __restrict__ bS1,
    const float* __restrict__ WS2, const float* __restrict__ bS2,
    const float* __restrict__ alpha_1,
    const float* __restrict__ W41, const float* __restrict__ b41,
    const float* __restrict__ W42, const float* __restrict__ b42,
    float* __restrict__ gOut) {
  __shared__ float sT[SEGN], sAvg[VV], sMax[VV], sAo[VV], sMo[VV], sOS[VV];
  __shared__ float sHa[10], sHm[10];
  const int b = blockIdx.x;
  const int tid = threadIdx.x;

  if (tid < SEGN) sT[tid] = ws_in[O_OUTT + b * SEGN + tid];
  __syncthreads();

  if (tid < VV) {
    float acc = 0.f, mx = -__builtin_inff();
    #pragma unroll
    for (int s = 0; s < SEGN; ++s) {
      const float o = sT[s];
      const int off = (b * SEGN + s) * VV + tid;
      acc += o * ws_in[O_SUMC + off];
      const float cand =
          (o >= 0.f) ? o * ws_in[O_MAXC + off] : o * ws_in[O_MINC + off];
      mx = fmaxf(mx, cand);
    }
    sAvg[tid] = acc * (1.0f / (float)(CC * SEGN));
    sMax[tid] = mx;
  }
  __syncthreads();

  if (tid < 10) {  // fcS hidden layer, both inputs
    float ha = bS1[tid], hm = bS1[tid];
    for (int v = 0; v < VV; ++v) {
      const float w = WS1[tid * VV + v];
      ha += w * sAvg[v];
      hm += w * sMax[v];
    }
    sHa[tid] = fmaxf(ha, 0.f);
    sHm[tid] = fmaxf(hm, 0.f);
  }
  __syncthreads();

  if (tid < VV) {  // fcS output layer
    float oa = bS2[tid], om = bS2[tid];
    #pragma unroll
    for (int i = 0; i < 10; ++i) {
      const float w = WS2[tid * 10 + i];
      oa += w * sHa[i];
      om += w * sHm[i];
    }
    sAo[tid] = oa;
    sMo[tid] = om;
    sOS[tid] = 0.5f * (oa + om);
  }
  __syncthreads();

  if (tid < VV) {  // Ak, feats, fc_4, sigmoid
    const float alpha = alpha_1[0];
    float f[5];
    #pragma unroll
    for (int k = 0; k < 3; ++k) {
      float acc = 0.f;
      const float* am = amat + k * VV * VV + tid;  // amat[k][v][w], w = tid
      for (int v = 0; v < VV; ++v) acc += sOS[v] * am[v * VV];
      f[k] = acc * alpha;
    }
    f[3] = sAo[tid];
    f[4] = sMo[tid];
    float acc = b42[0];
    #pragma unroll
    for (int c = 0; c < 8; ++c) {
      const float w41 = W41[c], bb = b41[c];
      #pragma unroll
      for (int k = 0; k < 5; ++k)
        acc += fmaxf(f[k] * w41 + bb, 0.f) * W42[c * 5 + k];
    }
    gOut[b * VV + tid] = 1.0f / (1.0f + __expf(-acc));
  }
}

// =====================================================================
// Kernel 5: out = x * outT[b,s] * g[b,v]. x re-read mostly hits the 192MB
// L2; stores are coalesced. 4 elements/thread, flat contiguous mapping.
// =====================================================================
__global__ void __launch_bounds__(256) scale_kernel(
    const float* __restrict__ x, const float* __restrict__ outT,
    const float* __restrict__ g, float* __restrict__ out) {
  const unsigned base = blockIdx.x * 1024u + threadIdx.x;
  #pragma unroll
  for (int j = 0; j < 4; ++j) {
    const unsigned idx = base + j * 256u;
    if (idx < (unsigned)TOT) {
      const unsigned v = idx % VV;
      const unsigned q = idx / VV;
      const unsigned t = q % TT;
      const unsigned n = (q / TT) / CC;
      const unsigned b = (t >> 4) * NN + n;
      out[idx] = x[idx] * outT[b * SEGN + (t & 15)] * g[b * VV + v];
    }
  }
}

// =====================================================================
extern "C" void kernel_launch(void* const* d_in, const int* in_sizes, int n_in,
                              void* d_out, int out_size, void* d_ws,
                              size_t ws_size, hipStream_t stream) {
  const float* x       = (const float*)d_in[0];
  const float* A       = (const float*)d_in[1];
  const float* alpha_1 = (const float*)d_in[2];
  const float* WT1 = (const float*)d_in[3];
  const float* bT1 = (const float*)d_in[4];
  const float* WT2 = (const float*)d_in[5];
  const float* bT2 = (const float*)d_in[6];
  const float* WS1 = (const float*)d_in[7];
  const float* bS1 = (const float*)d_in[8];
  const float* WS2 = (const float*)d_in[9];
  const float* bS2 = (const float*)d_in[10];
  const float* WA1 = (const float*)d_in[11];
  const float* bA1 = (const float*)d_in[12];
  const float* WA2 = (const float*)d_in[13];
  const float* bA2 = (const float*)d_in[14];
  const float* W41 = (const float*)d_in[15];
  const float* b41 = (const float*)d_in[16];
  const float* W42 = (const float*)d_in[17];
  const float* b42 = (const float*)d_in[18];
  float* out = (float*)d_out;
  float* ws = (float*)d_ws;

  // 1) pooling pass over x
  pool_kernel<<<BB * SEGN, 256, 0, stream>>>(x, ws);
  // 2) temporal MLP as WMMA GEMM (256x16 @ 16x16, two layers, avg+max)
  fcT_wmma_kernel<<<16, 32, 0, stream>>>(ws, WT1, bT1, WT2, bT2, ws + O_OUTT);
  // 3) pointwise a_mats
  amat_kernel<<<(3 * VV * VV + 255) / 256, 256, 0, stream>>>(
      A, WA1, bA1, WA2, bA2, ws + O_AMAT);
  // 4) spatial attention + gate per segment-batch
  spatial_kernel<<<BB, 64, 0, stream>>>(ws, ws + O_AMAT, WS1, bS1, WS2, bS2,
                                        alpha_1, W41, b41, W42, b42, ws + O_G);
  // 5) elementwise scaling pass
  scale_kernel<<<(TOT + 1023) / 1024, 256, 0, stream>>>(x, ws + O_OUTT,
                                                        ws + O_G, out);
}